// CatmullRomLayer_41729902248457
// MI455X (gfx1250) — compile-verified
//
#include <hip/hip_runtime.h>
#include <stdint.h>

typedef unsigned int uint;
typedef uint v4u __attribute__((ext_vector_type(4)));
typedef int  v8i __attribute__((ext_vector_type(8)));
typedef int  v4i __attribute__((ext_vector_type(4)));

#define KN 48          // knots per dim (N) -- constexpr so row offsets fold to immediates
#define NV 16          // output channels (V)
#define PPW 8          // points per wave (4 lanes each)

struct DimW { float w0, w1, w2, w3; int bs; };

// Catmull-Rom 4-tap weights + base index with linear-extrapolation padding
// folded into the weights so all grid indices stay in [0, KN-1].
__device__ inline DimW dim_weights(const float* kd, float xv) {
    // largest i in [0, KN-1] with kd[i] <= xv (branchless binary search)
    int lo = 0;
#pragma unroll
    for (int s = 32; s > 0; s >>= 1) {
        int cand = lo + s;
        if (cand <= KN - 1 && kd[cand] <= xv) lo = cand;
    }
    int idx = lo > KN - 2 ? KN - 2 : lo;

    float t0 = kd[idx], t1 = kd[idx + 1];
    float dt = t1 - t0;
    float u  = (xv - t0) / dt;
    float u2 = u * u, u3 = u2 * u;
    float h00 =  2.0f * u3 - 3.0f * u2 + 1.0f;
    float h10 =         u3 - 2.0f * u2 + u;
    float h01 = -2.0f * u3 + 3.0f * u2;
    float h11 =         u3 -        u2;

    float tpl = (idx == 0)      ? (2.0f * kd[0]      - kd[1])      : kd[idx - 1];
    float tph = (idx == KN - 2) ? (2.0f * kd[KN - 1] - kd[KN - 2]) : kd[idx + 2];
    float a  = dt / (t1  - tpl);
    float bb = dt / (tph - t0);

    float w0 = -h10 * a;
    float w1 =  h00 - h11 * bb;
    float w2 =  h01 + h10 * a;
    float w3 =  h11 * bb;

    DimW r;
    if (idx == 0) {               // taps hit padded index -1: v(-1)=2g0-g1
        r.bs = 0;
        r.w0 = w1 + 2.0f * w0; r.w1 = w2 - w0; r.w2 = w3; r.w3 = 0.0f;
    } else if (idx == KN - 2) {   // taps hit padded index KN: v(KN)=2g(KN-1)-g(KN-2)
        r.bs = KN - 4;
        r.w0 = 0.0f; r.w1 = w0; r.w2 = w1 - w3; r.w3 = w2 + 2.0f * w3;
    } else {
        r.bs = idx - 1;
        r.w0 = w0; r.w1 = w1; r.w2 = w2; r.w3 = w3;
    }
    return r;
}

__global__ __launch_bounds__(256) void catmullrom3d_kernel(
    const float* __restrict__ x,      // [B,3]
    const float* __restrict__ knots,  // [3,KN]
    const float* __restrict__ grid,   // [KN,KN,KN,NV]
    float* __restrict__ out,          // [B,NV]
    int B)
{
    __shared__ float sk[3 * KN];      // knots staged in LDS (TDM destination, offset 0)

    // ---- Tensor Data Mover: DMA 3*KN floats (knots) global -> LDS, wave 0 only ----
    if (threadIdx.x < 32) {
        uint64_t ga = (uint64_t)(uintptr_t)knots;
        const uint n0 = 3u * KN;      // 1D tensor of 3*KN fp32, tile = whole tensor
        v4u g0;
        g0.x = 1u;                                        // count=1, user mode
        g0.y = 0u;                                        // lds_addr = 0 (sk at LDS base)
        g0.z = (uint)(ga & 0xFFFFFFFFu);                  // global_addr[31:0]
        g0.w = (uint)((ga >> 32) & 0x01FFFFFFu) | (2u << 30); // global_addr[56:32], type=2
        v8i g1;
        g1[0] = (int)(2u << 16);      // workgroup_mask=0, data_size=2 (4 bytes)
        g1[1] = (int)(n0 << 16);      // tensor_dim0[15:0]
        g1[2] = (int)(1u << 16);      // tensor_dim0 hi = 0, tensor_dim1 = 1
        g1[3] = (int)(n0 << 16);      // tile_dim0 = 3*KN
        g1[4] = 1;                    // tile_dim1=1, tile_dim2=0
        g1[5] = (int)n0;              // tensor_dim0_stride
        g1[6] = 0;
        g1[7] = 0;
        v4i z4 = {0, 0, 0, 0};
        v8i z8 = {0, 0, 0, 0, 0, 0, 0, 0};
        __builtin_amdgcn_tensor_load_to_lds(g0, g1, z4, z4, z8, 0);
        __builtin_amdgcn_s_wait_tensorcnt(0);
    }
    // TDM's LDS write is invisible to the optimizer; escape sk's address
    // through opaque asm + memory clobber so reads of sk[] survive.
    {
        float* skp = &sk[0];
        asm volatile("" : "+v"(skp) : : "memory");
    }
    __syncthreads();

    const int lane = threadIdx.x & 31;
    const int pp   = lane >> 2;       // point within wave: 0..7
    const int s    = lane & 3;        // channel-quad slot: 0..3 (v = 4s..4s+3)
    const int wave = (int)((blockIdx.x * blockDim.x + threadIdx.x) >> 5);
    const int b    = wave * PPW + pp; // this lane's point
    if (b >= B) return;

    const float x0 = x[(size_t)b * 3 + 0];
    const float x1 = x[(size_t)b * 3 + 1];
    const float x2 = x[(size_t)b * 3 + 2];

    // per-lane (= per-point) weights; amortized across 8 points per wave
    DimW d0 = dim_weights(&sk[0],      x0);
    DimW d1 = dim_weights(&sk[KN],     x1);
    DimW d2 = dim_weights(&sk[2 * KN], x2);

    // Neighborhood base: all 64 (i,j,q) offsets are compile-time immediates.
    const float* pbase = grid + ((size_t)(d0.bs * KN + d1.bs) * KN + d2.bs) * NV;

    // Warm the first rows (i=0) before the weight-heavy address-independent math
#pragma unroll
    for (int q = 0; q < 2; ++q) {
        int m = s + 4 * q;            // line 0..7 -> rows 0..3, both halves
        int pj = m >> 1, half = m & 1;
        __builtin_prefetch(pbase + (pj * KN) * NV + half * 32, 0, 3);
    }

    const float wi[4] = { d0.w0, d0.w1, d0.w2, d0.w3 };
    const float wj[4] = { d1.w0, d1.w1, d1.w2, d1.w3 };
    const float wk[4] = { d2.w0, d2.w1, d2.w2, d2.w3 };

    // lane slot s reads the float4 at dwords 16q + 4s of each 64-float row:
    // k = q (compile-time per unrolled quarter), v = 4s..4s+3 (in-lane, no reduce)
    const float* lbase = pbase + s * 4;
    float4 acc; acc.x = 0.0f; acc.y = 0.0f; acc.z = 0.0f; acc.w = 0.0f;
#pragma unroll
    for (int i = 0; i < 4; ++i) {
#pragma unroll
        for (int j = 0; j < 4; ++j) {
            const float* r = lbase + ((i * KN + j) * KN) * NV;  // immediate offsets
            float wij = wi[i] * wj[j];
#pragma unroll
            for (int q = 0; q < 4; ++q) {
                float4 g = *(const float4*)(r + q * NV);        // 64B/point segments
                float c = wij * wk[q];
                acc.x = fmaf(c, g.x, acc.x);
                acc.y = fmaf(c, g.y, acc.y);
                acc.z = fmaf(c, g.z, acc.z);
                acc.w = fmaf(c, g.w, acc.w);
            }
        }
    }

    // all 32 lanes store float4: 8 points x 64B = 512B contiguous per wave
    *(float4*)(out + (size_t)b * NV + s * 4) = acc;
}

extern "C" void kernel_launch(void* const* d_in, const int* in_sizes, int n_in,
                              void* d_out, int out_size, void* d_ws, size_t ws_size,
                              hipStream_t stream) {
    const float* x     = (const float*)d_in[0];
    const float* knots = (const float*)d_in[1];
    const float* grid  = (const float*)d_in[2];
    float* out = (float*)d_out;

    int B = in_sizes[0] / 3;   // 131072

    int threads = 256;                               // 8 waves -> 64 points per block
    int waves   = (B + PPW - 1) / PPW;               // 8 points per wave
    int blocks  = (waves * 32 + threads - 1) / threads;
    catmullrom3d_kernel<<<blocks, threads, 0, stream>>>(x, knots, grid, out, B);
}